// LSTMModel_46325517255368
// MI455X (gfx1250) — compile-verified
//
#include <hip/hip_runtime.h>

// Problem constants (match reference)
#define VDIM   256
#define BSZ    128
#define TLEN   1024
#define FOURV  1024

typedef float v2f __attribute__((ext_vector_type(2)));
typedef float v8f __attribute__((ext_vector_type(8)));

// D = A(16x4 f32) * B(4x16 f32) + C(16x16 f32)   -- CDNA5 V_WMMA_F32_16X16X4_F32
static __device__ __forceinline__ v8f wmma4(v2f a, v2f b, v8f c) {
    return __builtin_amdgcn_wmma_f32_16x16x4_f32(
        /*neg_a=*/false, a, /*neg_b=*/false, b,
        /*c_mod=*/(short)0, c, /*reuse_a=*/false, /*reuse_b=*/false);
}

// ---------------------------------------------------------------------------
// C[M,N] = A[M,K=256] * W[N,K=256]^T + bias[N]
// One wave computes a 32x64 strip of C: 2 A-frags x 4 B-frags -> 8 WMMA tiles.
// Per K=4 step: 6 x b64 loads feed 8 WMMAs (A reused 4x, B reused 2x).
// A-frag layout (16x4 f32): lane = m + 16*(k/2), vgpr = k%2
// B-frag layout (4x16 f32): lane = n + 16*(k/2), vgpr = k%2  (rows of W, K-contig)
// C/D layout: vgpr r, lanes 0-15 -> M=r, lanes 16-31 -> M=8+r, N = lane%16
// ---------------------------------------------------------------------------
__global__ __launch_bounds__(256) void gemm_bias_kernel(
    const float* __restrict__ A, const float* __restrict__ W,
    const float* __restrict__ bias,
    float* __restrict__ C, int M, int N)
{
    const int K = VDIM;
    const int wid  = (blockIdx.x << 3) + (threadIdx.x >> 5);
    const int lane = threadIdx.x & 31;
    const int nStrips = N >> 6;                 // 64-wide strips in N
    const int mtile = wid / nStrips;            // 32-row tiles in M
    const int n0    = (wid % nStrips) << 6;
    if (mtile * 32 >= M) return;

    const int rl   = lane & 15;
    const int half = lane >> 4;
    const int koff = half << 1;                 // 0 or 2

    const float* arow0 = A + (size_t)(mtile * 32 + rl) * K;
    const float* arow1 = arow0 + (size_t)16 * K;
    const float* wrow  = W + (size_t)(n0 + rl) * K;

    v8f acc[2][4] = {};
    for (int k = 0; k < K; k += 4) {
        const int kk = k + koff;
        v2f a0 = *(const v2f*)(arow0 + kk);
        v2f a1 = *(const v2f*)(arow1 + kk);
#pragma unroll
        for (int j = 0; j < 4; ++j) {
            v2f b = *(const v2f*)(wrow + (size_t)(j * 16) * K + kk);
            acc[0][j] = wmma4(a0, b, acc[0][j]);
            acc[1][j] = wmma4(a1, b, acc[1][j]);
        }
    }

#pragma unroll
    for (int i = 0; i < 2; ++i) {
#pragma unroll
        for (int j = 0; j < 4; ++j) {
            const int n = n0 + j * 16 + rl;
            const float bv = bias[n];
#pragma unroll
            for (int r = 0; r < 8; ++r) {
                const int m = mtile * 32 + i * 16 + r + (half << 3);
                C[(size_t)m * N + n] = acc[i][j][r] + bv;
            }
        }
    }
}

// ---------------------------------------------------------------------------
// One LSTM timestep: gates = xp[:,t,:] + h_prev @ W_hh^T ; state update.
// 128 waves: mtile (8) x vtile (16). Each wave holds i/f/g/o 16x16 tiles for
// the SAME (m, v) coordinates -> pointwise LSTM math entirely in registers.
// ---------------------------------------------------------------------------
__global__ __launch_bounds__(256) void lstm_step_kernel(
    const float* __restrict__ Whh,    // [4V, V]
    const float* __restrict__ xp,     // [B, T, 4V] (includes b_ih + b_hh)
    const float* __restrict__ hprev,  // [B, V]
    const float* __restrict__ cprev,  // [B, V]
    float* __restrict__ hnext,        // [B, V]
    float* __restrict__ cnext,        // [B, V]
    float* __restrict__ hs,           // [B, T, V]
    int t)
{
    const int K = VDIM;
    const int wid  = (blockIdx.x << 3) + (threadIdx.x >> 5);   // 0..127
    const int lane = threadIdx.x & 31;
    const int mtile = wid >> 4;                                // 0..7
    const int v0    = (wid & 15) << 4;                         // 0..240
    const int rl   = lane & 15;
    const int half = lane >> 4;
    const int koff = half << 1;

    const float* arow = hprev + (size_t)(mtile * 16 + rl) * K;
    const float* wrow = Whh + (size_t)(v0 + rl) * K;

    v8f acc[4] = {};
    for (int k = 0; k < K; k += 4) {
        const int kk = k + koff;
        v2f a = *(const v2f*)(arow + kk);
#pragma unroll
        for (int g = 0; g < 4; ++g) {          // gate order i, f, g, o
            v2f b = *(const v2f*)(wrow + (size_t)(g * VDIM) * K + kk);
            acc[g] = wmma4(a, b, acc[g]);
        }
    }

    const int v = v0 + rl;
#pragma unroll
    for (int r = 0; r < 8; ++r) {
        const int m = mtile * 16 + r + (half << 3);
        const float* xr = xp + ((size_t)m * TLEN + t) * FOURV;
        const float gi = acc[0][r] + xr[v];
        const float gf = acc[1][r] + xr[VDIM + v];
        const float gg = acc[2][r] + xr[2 * VDIM + v];
        const float go = acc[3][r] + xr[3 * VDIM + v];
        const float si = 1.0f / (1.0f + expf(-gi));
        const float sf = 1.0f / (1.0f + expf(-gf));
        const float tg = tanhf(gg);
        const float so = 1.0f / (1.0f + expf(-go));
        const size_t sidx = (size_t)m * VDIM + v;
        const float c = sf * cprev[sidx] + si * tg;
        const float h = so * tanhf(c);
        cnext[sidx] = c;
        hnext[sidx] = h;
        hs[((size_t)m * TLEN + t) * VDIM + v] = h;
    }
}

// ---------------------------------------------------------------------------
__global__ void zero_kernel(float* __restrict__ p, int n) {
    const int i = blockIdx.x * blockDim.x + threadIdx.x;
    if (i < n) p[i] = 0.0f;
}

// o[i] = a[i] + b[i]  (combine b_ih + b_hh once, so GEMM epilogue is branch-free)
__global__ void add_vec_kernel(const float* __restrict__ a,
                               const float* __restrict__ b,
                               float* __restrict__ o, int n) {
    const int i = blockIdx.x * blockDim.x + threadIdx.x;
    if (i < n) o[i] = a[i] + b[i];
}

// L2-normalize each row of fc_w (256x256). One thread per row.
__global__ __launch_bounds__(256) void normalize_rows_kernel(
    const float* __restrict__ w, float* __restrict__ wn)
{
    const int row = threadIdx.x;                 // 0..255, single block
    const float* wr = w + (size_t)row * VDIM;
    float s = 0.0f;
    for (int k = 0; k < VDIM; ++k) { const float x = wr[k]; s += x * x; }
    const float inv = 1.0f / sqrtf(s);
    float* on = wn + (size_t)row * VDIM;
    for (int k = 0; k < VDIM; ++k) on[k] = wr[k] * inv;
}

// ---------------------------------------------------------------------------
extern "C" void kernel_launch(void* const* d_in, const int* in_sizes, int n_in,
                              void* d_out, int out_size, void* d_ws, size_t ws_size,
                              hipStream_t stream)
{
    const float* x     = (const float*)d_in[0];
    const float* W_ih1 = (const float*)d_in[1];
    const float* W_hh1 = (const float*)d_in[2];
    const float* b_ih1 = (const float*)d_in[3];
    const float* b_hh1 = (const float*)d_in[4];
    const float* W_ih2 = (const float*)d_in[5];
    const float* W_hh2 = (const float*)d_in[6];
    const float* b_ih2 = (const float*)d_in[7];
    const float* b_hh2 = (const float*)d_in[8];
    const float* fc_w  = (const float*)d_in[9];
    const float* fc_b  = (const float*)d_in[10];

    // Workspace layout
    char* ws = (char*)d_ws;
    const size_t xp_bytes = (size_t)BSZ * TLEN * FOURV * sizeof(float);  // 512 MB
    const size_t hs_bytes = (size_t)BSZ * TLEN * VDIM * sizeof(float);   // 128 MB
    float* xp = (float*)ws;
    float* hs = (float*)(ws + xp_bytes);
    float* st = (float*)(ws + xp_bytes + hs_bytes);
    float* h0 = st;
    float* h1 = st + (size_t)BSZ * VDIM;
    float* c0 = st + (size_t)2 * BSZ * VDIM;
    float* c1 = st + (size_t)3 * BSZ * VDIM;
    float* wn = st + (size_t)4 * BSZ * VDIM;     // 256*256 floats
    float* bias1 = wn + (size_t)VDIM * VDIM;     // 1024 floats
    float* bias2 = bias1 + FOURV;                // 1024 floats

    const int M = BSZ * TLEN;                    // 131072 rows
    const int stateN = 4 * BSZ * VDIM;           // ping-pong h/c

    // Pre-combined gate biases
    add_vec_kernel<<<FOURV / 256, 256, 0, stream>>>(b_ih1, b_hh1, bias1, FOURV);
    add_vec_kernel<<<FOURV / 256, 256, 0, stream>>>(b_ih2, b_hh2, bias2, FOURV);

    // ---- Layer 1 ----
    {   // xp = x @ W_ih1^T + (b_ih1 + b_hh1)
        const int waves = (M / 32) * (FOURV / 64);
        gemm_bias_kernel<<<waves / 8, 256, 0, stream>>>(x, W_ih1, bias1, xp, M, FOURV);
    }
    zero_kernel<<<stateN / 256, 256, 0, stream>>>(st, stateN);
    for (int t = 0; t < TLEN; ++t) {
        const float* hp = (t & 1) ? h1 : h0;  float* hn = (t & 1) ? h0 : h1;
        const float* cp = (t & 1) ? c1 : c0;  float* cn = (t & 1) ? c0 : c1;
        lstm_step_kernel<<<16, 256, 0, stream>>>(W_hh1, xp, hp, cp, hn, cn, hs, t);
    }

    // ---- Layer 2 ----
    {   // xp = h1s @ W_ih2^T + (b_ih2 + b_hh2)  (reads hs before it is reused)
        const int waves = (M / 32) * (FOURV / 64);
        gemm_bias_kernel<<<waves / 8, 256, 0, stream>>>(hs, W_ih2, bias2, xp, M, FOURV);
    }
    zero_kernel<<<stateN / 256, 256, 0, stream>>>(st, stateN);
    for (int t = 0; t < TLEN; ++t) {
        const float* hp = (t & 1) ? h1 : h0;  float* hn = (t & 1) ? h0 : h1;
        const float* cp = (t & 1) ? c1 : c0;  float* cn = (t & 1) ? c0 : c1;
        lstm_step_kernel<<<16, 256, 0, stream>>>(W_hh2, xp, hp, cp, hn, cn, hs, t);
    }

    // ---- Normalized linear head ----
    normalize_rows_kernel<<<1, 256, 0, stream>>>(fc_w, wn);
    {   // out = h2s @ w_n^T + fc_b
        const int waves = (M / 32) * (VDIM / 64);
        gemm_bias_kernel<<<waves / 8, 256, 0, stream>>>(hs, wn, fc_b,
                                                        (float*)d_out, M, VDIM);
    }
}